// PathwayGNNLayer_37503654429371
// MI455X (gfx1250) — compile-verified
//
#include <hip/hip_runtime.h>
#include <hip/hip_bf16.h>
#include <math.h>

// Problem constants (match reference)
#define NUM_SRC 100000
#define NUM_DST 50000
#define NUM_E   800000
#define DIM     128

typedef __attribute__((ext_vector_type(2))) float v2f;
typedef __attribute__((ext_vector_type(8))) float v8f;

// ---------------------------------------------------------------------------
// Kernel 1: degree counts (f32 atomics; counts are exactly representable)
// ---------------------------------------------------------------------------
__global__ __launch_bounds__(256) void deg_kernel(const int* __restrict__ src,
                                                  const int* __restrict__ dst,
                                                  float* __restrict__ deg_src,
                                                  float* __restrict__ deg_dst) {
    int e = blockIdx.x * blockDim.x + threadIdx.x;
    if (e < NUM_E) {
        atomicAdd(&deg_src[src[e]], 1.0f);
        atomicAdd(&deg_dst[dst[e]], 1.0f);
    }
}

// ---------------------------------------------------------------------------
// Kernel 2: edge scatter. 32 lanes per edge, float4 per lane (128 feats).
// agg[dst[e]] += h[src[e]] * rsqrt(max(deg_src[src[e]],1))
// h rows (51 MB working set) + agg (25.6 MB) are L2-resident -> L2-bound.
// ---------------------------------------------------------------------------
__global__ __launch_bounds__(256) void scatter_kernel(const float* __restrict__ h,
                                                      const int* __restrict__ src,
                                                      const int* __restrict__ dst,
                                                      const float* __restrict__ deg_src,
                                                      float* __restrict__ agg) {
    int t    = blockIdx.x * blockDim.x + threadIdx.x;
    int e    = t >> 5;          // edge index (32 lanes / edge)
    int lane = t & 31;          // feature quad index
    if (e >= NUM_E) return;

    int s = src[e];
    int d = dst[e];
    float sc = rsqrtf(fmaxf(deg_src[s], 1.0f));

    const float4* hp = (const float4*)(h + (size_t)s * DIM);
    float4 v = hp[lane];

    float* ap = agg + (size_t)d * DIM + lane * 4;
    atomicAdd(ap + 0, v.x * sc);
    atomicAdd(ap + 1, v.y * sc);
    atomicAdd(ap + 2, v.z * sc);
    atomicAdd(ap + 3, v.w * sc);
}

// ---------------------------------------------------------------------------
// Kernel 3: fused  out = (agg * rsqrt(deg_dst)) @ W + b,
//                  alpha = sigmoid(out @ w_attn),  out_scaled = out * alpha.
// One block = 16 rows; 8 waves each compute a 16x16 tile via 32x
// V_WMMA_F32_16X16X4_F32 (full fp32 precision; GEMM is not the bottleneck).
// ---------------------------------------------------------------------------
#define LDS_STRIDE 132   // 128 + 4: spreads the column-broadcast A reads over banks

__global__ __launch_bounds__(256) void gemm_attn_kernel(const float* __restrict__ agg,
                                                        const float* __restrict__ deg_dst,
                                                        const float* __restrict__ W,
                                                        const float* __restrict__ bias,
                                                        const float* __restrict__ w_attn,
                                                        float* __restrict__ out_scaled,
                                                        float* __restrict__ alpha_out) {
    __shared__ float As[16 * LDS_STRIDE];   // scaled A tile (16 x 128)
    __shared__ float Os[16 * LDS_STRIDE];   // out tile before attention scaling
    __shared__ float alphaS[16];

    const int row0 = blockIdx.x * 16;

    // Stage scaled A tile into LDS (256 threads x 8 elems)
    for (int i = threadIdx.x; i < 16 * DIM; i += 256) {
        int r = i >> 7;          // 0..15
        int c = i & 127;         // 0..127
        int gr = row0 + r;
        float val = 0.0f;
        if (gr < NUM_DST) {
            val = agg[(size_t)gr * DIM + c] * rsqrtf(fmaxf(deg_dst[gr], 1.0f));
        }
        As[r * LDS_STRIDE + c] = val;
    }
    __syncthreads();

    const int wave    = threadIdx.x >> 5;    // 0..7 -> column tile of W
    const int lane    = threadIdx.x & 31;
    const int colBase = wave * 16;
    const int nn      = lane & 15;           // A row (M) / B col (N)
    const int kHalf   = (lane >> 4) * 2;     // K sub-pair: lanes 0-15 -> K{0,1}, 16-31 -> K{2,3}

    v8f acc = {};
    #pragma unroll
    for (int k0 = 0; k0 < DIM; k0 += 4) {
        v2f a, bf;
        // A fragment: lane L holds A[M = L%16][k0 + (L/16)*2 + {0,1}]
        a.x = As[nn * LDS_STRIDE + k0 + kHalf];
        a.y = As[nn * LDS_STRIDE + k0 + kHalf + 1];
        // B fragment: lane L holds W[k0 + (L/16)*2 + {0,1}][colBase + L%16]
        bf.x = W[(size_t)(k0 + kHalf) * DIM + colBase + nn];
        bf.y = W[(size_t)(k0 + kHalf + 1) * DIM + colBase + nn];
        // D = A*B + C, full fp32  -> v_wmma_f32_16x16x4_f32
        acc = __builtin_amdgcn_wmma_f32_16x16x4_f32(
            /*neg_a=*/false, a, /*neg_b=*/false, bf,
            /*c_mod=*/(short)0, acc, /*reuse_a=*/false, /*reuse_b=*/false);
    }

    // Spill C/D fragment (+bias) to LDS: lane L, vgpr v -> row v+8*(L/16), col colBase+L%16
    const float bcol = bias[colBase + nn];
    #pragma unroll
    for (int v = 0; v < 8; ++v) {
        int r = v + 8 * (lane >> 4);
        Os[r * LDS_STRIDE + colBase + nn] = acc[v] + bcol;
    }
    __syncthreads();

    // Attention: one thread per row computes dot(out_row, w_attn), sigmoid
    if (threadIdx.x < 16) {
        int r = threadIdx.x;
        float dot = 0.0f;
        #pragma unroll 8
        for (int k = 0; k < DIM; ++k) dot += Os[r * LDS_STRIDE + k] * w_attn[k];
        float al = 1.0f / (1.0f + __expf(-dot));
        alphaS[r] = al;
        int gr = row0 + r;
        if (gr < NUM_DST) alpha_out[gr] = al;
    }
    __syncthreads();

    // Fused writeback: out * alpha
    for (int i = threadIdx.x; i < 16 * DIM; i += 256) {
        int r = i >> 7;
        int c = i & 127;
        int gr = row0 + r;
        if (gr < NUM_DST) {
            out_scaled[(size_t)gr * DIM + c] = Os[r * LDS_STRIDE + c] * alphaS[r];
        }
    }
}

// ---------------------------------------------------------------------------
// Launch
// ---------------------------------------------------------------------------
extern "C" void kernel_launch(void* const* d_in, const int* in_sizes, int n_in,
                              void* d_out, int out_size, void* d_ws, size_t ws_size,
                              hipStream_t stream) {
    const float* h      = (const float*)d_in[0];   // [NUM_SRC, 128]
    const int*   src    = (const int*)d_in[1];     // [E]
    const int*   dst    = (const int*)d_in[2];     // [E]
    const float* W      = (const float*)d_in[3];   // [128, 128]
    const float* bias   = (const float*)d_in[4];   // [128]
    const float* w_attn = (const float*)d_in[5];   // [128]

    float* out   = (float*)d_out;                       // [NUM_DST*128] out*alpha
    float* alpha = out + (size_t)NUM_DST * DIM;         // [NUM_DST] alpha

    // Workspace: agg [NUM_DST*128] | deg_src [NUM_SRC] | deg_dst [NUM_DST]
    float* agg     = (float*)d_ws;
    float* deg_src = agg + (size_t)NUM_DST * DIM;
    float* deg_dst = deg_src + NUM_SRC;
    size_t zero_bytes = ((size_t)NUM_DST * DIM + NUM_SRC + NUM_DST) * sizeof(float);
    hipMemsetAsync(d_ws, 0, zero_bytes, stream);

    deg_kernel<<<(NUM_E + 255) / 256, 256, 0, stream>>>(src, dst, deg_src, deg_dst);

    scatter_kernel<<<((size_t)NUM_E * 32 + 255) / 256, 256, 0, stream>>>(
        h, src, dst, deg_src, agg);

    gemm_attn_kernel<<<(NUM_DST + 15) / 16, 256, 0, stream>>>(
        agg, deg_dst, W, bias, w_attn, out, alpha);
}